// MockBoxLoss_34686155883099
// MI455X (gfx1250) — compile-verified
//
#include <hip/hip_runtime.h>
#include <math.h>

#define CIOU_EPS 1e-9f
#define FOUR_OVER_PI2 0.4052847345693511f

typedef __attribute__((ext_vector_type(2))) float v2f;
typedef __attribute__((ext_vector_type(8))) float v8f;

// ds_swizzle_b32 group-of-32 xor pattern: offset[14:10]=xor, [9:5]=or, [4:0]=and.
// With and=0x1F, or=0: new_lane = lane ^ xor_mask. Immediate-only, no address VALU.
#define SWZ_XOR(x, k) \
    __int_as_float(__builtin_amdgcn_ds_swizzle(__float_as_int(x), ((k) << 10) | 0x1F))

// Exact fp32 wave32 sum using V_WMMA_F32_16X16X4_F32:
//   A = ones(16x4), B vgpr0 = lane partials p, B vgpr1 = 0
//   => D[m][n] = B[0][n] + B[2][n] = p[n] + p[n+16]  (exact fp32 MAC, RNE)
// D vgpr0 holds t[n] = p[n]+p[n+16] replicated in both lane halves; four
// immediate ds_swizzle xor-folds finish the 16-way sum. Requires EXEC all-1s.
__device__ __forceinline__ float wave_sum32(float p) {
    v2f a; a[0] = 1.0f; a[1] = 1.0f;   // A: ones
    v2f b; b[0] = p;    b[1] = 0.0f;   // B: partials in K=0 (lanes 0-15) / K=2 (lanes 16-31)
    v8f c = {};
    c = __builtin_amdgcn_wmma_f32_16x16x4_f32(
        /*neg_a=*/false, a, /*neg_b=*/false, b,
        /*c_mod=*/(short)0, c, /*reuse_a=*/false, /*reuse_b=*/false);
    float t = c[0];
    t += SWZ_XOR(t, 1);
    t += SWZ_XOR(t, 2);
    t += SWZ_XOR(t, 4);
    t += SWZ_XOR(t, 8);
    return t;   // every lane holds the full 32-lane sum
}

__global__ void __launch_bounds__(256)
ciou_partial_kernel(const float* __restrict__ pb,
                    const float* __restrict__ tb,
                    const unsigned char* __restrict__ mask,
                    const float* __restrict__ box_norm,
                    float* __restrict__ partials,
                    int n) {
    __shared__ float lds[8];
    const int i = blockIdx.x * blockDim.x + threadIdx.x;

    float acc = 0.0f;
    if (i < n) {
        const float4 A = ((const float4*)pb)[i];   // x1a,y1a,x2a,y2a
        const float4 B = ((const float4*)tb)[i];   // x1b,y1b,x2b,y2b

        const float x1a = A.x, y1a = A.y, x2a = A.z, y2a = A.w;
        const float x1b = B.x, y1b = B.y, x2b = B.z, y2b = B.w;

        const float xmin = fmaxf(x1a, x1b), ymin = fmaxf(y1a, y1b);
        const float xmax = fminf(x2a, x2b), ymax = fminf(y2a, y2b);
        const float inter = fmaxf(xmax - xmin, 0.0f) * fmaxf(ymax - ymin, 0.0f);

        const float wa = x2a - x1a, ha = y2a - y1a;
        const float wb = x2b - x1b, hb = y2b - y1b;
        const float area_a = wa * ha, area_b = wb * hb;
        const float uni = area_a + area_b - inter;
        const float iou = inter / (uni + CIOU_EPS);

        const float cx = (x1a + x2a - x1b - x2b) * 0.5f;
        const float cy = (y1a + y2a - y1b - y2b) * 0.5f;
        const float cent2 = cx * cx + cy * cy;

        const float cw = fmaxf(x2a, x2b) - fminf(x1a, x1b);
        const float ch = fmaxf(y2a, y2b) - fminf(y1a, y1b);
        const float diag2 = cw * cw + ch * ch;

        const float diou = iou - cent2 / (diag2 + CIOU_EPS);

        const float at = atanf(wa / (ha + CIOU_EPS)) - atanf(wb / (hb + CIOU_EPS));
        const float v  = FOUR_OVER_PI2 * at * at;
        const float alpha = v / (v - iou + 1.0f + CIOU_EPS);
        const float ciou  = diou - alpha * v;

        const float m = mask[i] ? 1.0f : 0.0f;
        acc = (1.0f - ciou) * m * box_norm[i];
    }

    // EXEC is all-ones here (no early returns) -> safe for WMMA + swizzle.
    const float wsum = wave_sum32(acc);
    const int wave = threadIdx.x >> 5;
    const int lane = threadIdx.x & 31;
    if (lane == 0) lds[wave] = wsum;
    __syncthreads();
    if (threadIdx.x == 0) {
        float s = 0.0f;
#pragma unroll
        for (int w = 0; w < 8; ++w) s += lds[w];
        partials[blockIdx.x] = s;
    }
}

__global__ void __launch_bounds__(256)
finalize_kernel(const float* __restrict__ partials, int nparts,
                const float* __restrict__ cls_norm,
                float* __restrict__ out) {
    __shared__ float lds[8];
    float acc = 0.0f;
    for (int i = threadIdx.x; i < nparts; i += 256) acc += partials[i];

    // Threads reconverge here; EXEC all ones for the WMMA fold.
    const float wsum = wave_sum32(acc);
    const int wave = threadIdx.x >> 5;
    const int lane = threadIdx.x & 31;
    if (lane == 0) lds[wave] = wsum;
    __syncthreads();
    if (threadIdx.x == 0) {
        float s = 0.0f;
#pragma unroll
        for (int w = 0; w < 8; ++w) s += lds[w];
        out[0] = s / cls_norm[0];
    }
}

extern "C" void kernel_launch(void* const* d_in, const int* in_sizes, int n_in,
                              void* d_out, int out_size, void* d_ws, size_t ws_size,
                              hipStream_t stream) {
    const float*         pb       = (const float*)d_in[0];         // (B,A,4) f32
    const float*         tb       = (const float*)d_in[1];         // (B,A,4) f32
    const unsigned char* mask     = (const unsigned char*)d_in[2]; // (B,A) bool
    const float*         box_norm = (const float*)d_in[3];         // (N,) f32
    const float*         cls_norm = (const float*)d_in[4];         // scalar f32

    const int n = in_sizes[2];                 // number of boxes = B*A
    const int threads = 256;
    const int nblocks = (n + threads - 1) / threads;   // 525 for N=134400

    float* partials = (float*)d_ws;

    ciou_partial_kernel<<<nblocks, threads, 0, stream>>>(pb, tb, mask, box_norm, partials, n);
    finalize_kernel<<<1, threads, 0, stream>>>(partials, nblocks, cls_norm, (float*)d_out);
}